// RnnSenderGS_7095285973733
// MI455X (gfx1250) — compile-verified
//
#include <hip/hip_runtime.h>
#include <math.h>
#include <cfloat>

// Problem dims (fixed by reference)
#define V_   2048
#define E_   512
#define H_   1024
#define T_   16
#define B_   512
#define DIN_ 1024

typedef __attribute__((ext_vector_type(2))) float v2f;
typedef __attribute__((ext_vector_type(8))) float v8f;

__device__ __forceinline__ v8f wmma4(v2f a, v2f b, v8f c) {
    // D = A(16x4 f32) * B(4x16 f32) + C(16x16 f32)
    return __builtin_amdgcn_wmma_f32_16x16x4_f32(
        /*neg_a=*/false, a, /*neg_b=*/false, b,
        /*c_mod=*/(short)0, c, /*reuse_a=*/false, /*reuse_b=*/false);
}

__device__ __forceinline__ float sigmoidf_(float x) {
    return 1.0f / (1.0f + expf(-x));
}

__device__ __forceinline__ float gumbelf_(float u) {
    u = fminf(fmaxf(u, 1e-7f), 1.0f - 1e-7f);
    return -logf(-logf(u));
}

// ---------------------------------------------------------------------------
// Generic C[M,N] = A[M,K] @ W[N,K]^T + bias[N].  A rows strided by lda.
// One wave computes a 16(M) x 64(N) strip: 4 WMMA tiles sharing the A operand
// (1 A-load + 4 B-loads per K-step -> 6.4 flop/byte of cache traffic).
// ---------------------------------------------------------------------------
template <int M, int N, int K>
__global__ void __launch_bounds__(256)
gemm_bias_kernel(const float* __restrict__ A, int lda,
                 const float* __restrict__ W,
                 const float* __restrict__ bias,
                 float* __restrict__ C) {
    constexpr int TN = N / 64;  // N-strips of 64
    const int lane = threadIdx.x & 31;
    const int wave = threadIdx.x >> 5;
    const int strip = blockIdx.x * 8 + wave;
    const int tm = strip / TN;
    const int tn = strip % TN;
    const int idx = lane & 15;
    const int half = lane >> 4;

    const float* __restrict__ pa = A + (size_t)(tm * 16 + idx) * lda + 2 * half;
    const float* __restrict__ pw0 = W + (size_t)(tn * 64 +  0 + idx) * K + 2 * half;
    const float* __restrict__ pw1 = W + (size_t)(tn * 64 + 16 + idx) * K + 2 * half;
    const float* __restrict__ pw2 = W + (size_t)(tn * 64 + 32 + idx) * K + 2 * half;
    const float* __restrict__ pw3 = W + (size_t)(tn * 64 + 48 + idx) * K + 2 * half;

    v8f acc0 = {0.f,0.f,0.f,0.f,0.f,0.f,0.f,0.f};
    v8f acc1 = acc0, acc2 = acc0, acc3 = acc0;
    for (int k = 0; k < K; k += 4) {
        v2f a = *(const v2f*)(pa + k);
        acc0 = wmma4(a, *(const v2f*)(pw0 + k), acc0);
        acc1 = wmma4(a, *(const v2f*)(pw1 + k), acc1);
        acc2 = wmma4(a, *(const v2f*)(pw2 + k), acc2);
        acc3 = wmma4(a, *(const v2f*)(pw3 + k), acc3);
    }

    const int row0 = tm * 16 + half * 8;
    v8f accs[4] = {acc0, acc1, acc2, acc3};
#pragma unroll
    for (int j = 0; j < 4; ++j) {
        const int col = tn * 64 + j * 16 + idx;
        const float bv = bias[col];
#pragma unroll
        for (int r = 0; r < 8; ++r)
            C[(size_t)(row0 + r) * N + col] = accs[j][r] + bv;
    }
}

// ---------------------------------------------------------------------------
// Fused LSTM step: gates = e @ W_ih^T + h @ W_hh^T + b_ih + b_hh, then
// c' = sig(f)*c + sig(i)*tanh(g);  h' = sig(o)*tanh(c').
// One wave computes a 32(M) x 16(N) tile of ALL FOUR gates: per K-step,
// 2 A-loads + 4 B-loads feed 8 WMMAs (10.7 flop/byte of cache traffic).
// ---------------------------------------------------------------------------
__global__ void __launch_bounds__(256)
lstm_step_kernel(const float* __restrict__ e,     // [B,E]
                 const float* __restrict__ h_in,  // [B,H]
                 const float* __restrict__ c_in,  // [B,H]
                 const float* __restrict__ W_ih,  // [4H,E]
                 const float* __restrict__ W_hh,  // [4H,H]
                 const float* __restrict__ b_ih,  // [4H]
                 const float* __restrict__ b_hh,  // [4H]
                 float* __restrict__ h_out,       // [B,H]
                 float* __restrict__ c_out) {     // [B,H]
    constexpr int TN = H_ / 16;  // 64
    const int lane = threadIdx.x & 31;
    const int wave = threadIdx.x >> 5;
    const int tile = blockIdx.x * 8 + wave;
    const int tm = tile / TN;                // M-block of 32 rows
    const int tn = tile % TN;
    const int idx = lane & 15;
    const int half = lane >> 4;
    const int col = tn * 16 + idx;

    v8f z = {0.f,0.f,0.f,0.f,0.f,0.f,0.f,0.f};
    v8f ai0 = z, af0 = z, ag0 = z, ao0 = z;   // rows tm*32 .. +15
    v8f ai1 = z, af1 = z, ag1 = z, ao1 = z;   // rows tm*32+16 .. +31

    {   // input contribution: K = E
        const float* pa0 = e + (size_t)(tm * 32 +  0 + idx) * E_ + 2 * half;
        const float* pa1 = e + (size_t)(tm * 32 + 16 + idx) * E_ + 2 * half;
        const float* p0 = W_ih + (size_t)(0 * H_ + col) * E_ + 2 * half;
        const float* p1 = W_ih + (size_t)(1 * H_ + col) * E_ + 2 * half;
        const float* p2 = W_ih + (size_t)(2 * H_ + col) * E_ + 2 * half;
        const float* p3 = W_ih + (size_t)(3 * H_ + col) * E_ + 2 * half;
        for (int k = 0; k < E_; k += 4) {
            v2f a0 = *(const v2f*)(pa0 + k);
            v2f a1 = *(const v2f*)(pa1 + k);
            v2f b0 = *(const v2f*)(p0 + k);
            v2f b1 = *(const v2f*)(p1 + k);
            v2f b2 = *(const v2f*)(p2 + k);
            v2f b3 = *(const v2f*)(p3 + k);
            ai0 = wmma4(a0, b0, ai0);  ai1 = wmma4(a1, b0, ai1);
            af0 = wmma4(a0, b1, af0);  af1 = wmma4(a1, b1, af1);
            ag0 = wmma4(a0, b2, ag0);  ag1 = wmma4(a1, b2, ag1);
            ao0 = wmma4(a0, b3, ao0);  ao1 = wmma4(a1, b3, ao1);
        }
    }
    {   // recurrent contribution: K = H
        const float* pa0 = h_in + (size_t)(tm * 32 +  0 + idx) * H_ + 2 * half;
        const float* pa1 = h_in + (size_t)(tm * 32 + 16 + idx) * H_ + 2 * half;
        const float* p0 = W_hh + (size_t)(0 * H_ + col) * H_ + 2 * half;
        const float* p1 = W_hh + (size_t)(1 * H_ + col) * H_ + 2 * half;
        const float* p2 = W_hh + (size_t)(2 * H_ + col) * H_ + 2 * half;
        const float* p3 = W_hh + (size_t)(3 * H_ + col) * H_ + 2 * half;
        for (int k = 0; k < H_; k += 4) {
            v2f a0 = *(const v2f*)(pa0 + k);
            v2f a1 = *(const v2f*)(pa1 + k);
            v2f b0 = *(const v2f*)(p0 + k);
            v2f b1 = *(const v2f*)(p1 + k);
            v2f b2 = *(const v2f*)(p2 + k);
            v2f b3 = *(const v2f*)(p3 + k);
            ai0 = wmma4(a0, b0, ai0);  ai1 = wmma4(a1, b0, ai1);
            af0 = wmma4(a0, b1, af0);  af1 = wmma4(a1, b1, af1);
            ag0 = wmma4(a0, b2, ag0);  ag1 = wmma4(a1, b2, ag1);
            ao0 = wmma4(a0, b3, ao0);  ao1 = wmma4(a1, b3, ao1);
        }
    }

    const float bi = b_ih[0 * H_ + col] + b_hh[0 * H_ + col];
    const float bf = b_ih[1 * H_ + col] + b_hh[1 * H_ + col];
    const float bg = b_ih[2 * H_ + col] + b_hh[2 * H_ + col];
    const float bo = b_ih[3 * H_ + col] + b_hh[3 * H_ + col];

    v8f gi[2] = {ai0, ai1}, gf[2] = {af0, af1};
    v8f gg[2] = {ag0, ag1}, go[2] = {ao0, ao1};
#pragma unroll
    for (int mh = 0; mh < 2; ++mh) {
        const int row0 = tm * 32 + mh * 16 + half * 8;
#pragma unroll
        for (int r = 0; r < 8; ++r) {
            const int b = row0 + r;
            const float iv = sigmoidf_(gi[mh][r] + bi);
            const float fv = sigmoidf_(gf[mh][r] + bf);
            const float gv = tanhf(gg[mh][r] + bg);
            const float ov = sigmoidf_(go[mh][r] + bo);
            const float cn = fv * c_in[(size_t)b * H_ + col] + iv * gv;
            const float hn = ov * tanhf(cn);
            c_out[(size_t)b * H_ + col] = cn;
            h_out[(size_t)b * H_ + col] = hn;
        }
    }
}

// ---------------------------------------------------------------------------
// Per-row gumbel softmax / argmax / log-softmax / entropy. One block per row.
// ---------------------------------------------------------------------------
__device__ __forceinline__ float block_max_(float v, float* red) {
    const int tid = threadIdx.x;
    red[tid] = v; __syncthreads();
    for (int s = 128; s > 0; s >>= 1) {
        if (tid < s) red[tid] = fmaxf(red[tid], red[tid + s]);
        __syncthreads();
    }
    float r = red[0]; __syncthreads();
    return r;
}

__device__ __forceinline__ float block_sum_(float v, float* red) {
    const int tid = threadIdx.x;
    red[tid] = v; __syncthreads();
    for (int s = 128; s > 0; s >>= 1) {
        if (tid < s) red[tid] += red[tid + s];
        __syncthreads();
    }
    float r = red[0]; __syncthreads();
    return r;
}

__global__ void __launch_bounds__(256)
sample_kernel(const float* __restrict__ logits,  // [B,V]
              const float* __restrict__ u_con,   // [T,B,V]
              const float* __restrict__ u_rel,
              const float* __restrict__ u_cat,
              float* __restrict__ seq,           // [B,T+1,V]
              float* __restrict__ ohs,           // [B,T+1,V]
              float* __restrict__ lps,           // [B,T,V]
              float* __restrict__ ent,           // [T,B]
              int t) {
    __shared__ float red[256];
    __shared__ float redv[256];
    __shared__ int   redi[256];
    constexpr int PER = V_ / 256;  // 8
    const int b = blockIdx.x;
    const int tid = threadIdx.x;
    const float* lrow = logits + (size_t)b * V_;
    const size_t noff = ((size_t)t * B_ + b) * V_;

    float l[PER], g1[PER], g2[PER], g3[PER];
#pragma unroll
    for (int i = 0; i < PER; ++i) {
        const int v = tid + i * 256;
        l[i]  = lrow[v];
        g1[i] = gumbelf_(u_con[noff + v]);
        g2[i] = gumbelf_(u_rel[noff + v]);
        g3[i] = gumbelf_(u_cat[noff + v]);
    }

    // --- sample_concrete = softmax(logits + g1)  (TAU == 1) ---
    float m = -FLT_MAX;
#pragma unroll
    for (int i = 0; i < PER; ++i) m = fmaxf(m, l[i] + g1[i]);
    m = block_max_(m, red);
    float s = 0.f, e1[PER];
#pragma unroll
    for (int i = 0; i < PER; ++i) { e1[i] = expf(l[i] + g1[i] - m); s += e1[i]; }
    s = block_sum_(s, red);
    {
        const float inv = 1.0f / s;
        float* seqrow = seq + ((size_t)b * (T_ + 1) + t) * V_;
#pragma unroll
        for (int i = 0; i < PER; ++i) seqrow[tid + i * 256] = e1[i] * inv;
    }

    // --- logits_relaxed = softmax(logits + g2) ---
    float m2 = -FLT_MAX;
#pragma unroll
    for (int i = 0; i < PER; ++i) m2 = fmaxf(m2, l[i] + g2[i]);
    m2 = block_max_(m2, red);
    float s2 = 0.f, lr[PER];
#pragma unroll
    for (int i = 0; i < PER; ++i) { lr[i] = expf(l[i] + g2[i] - m2); s2 += lr[i]; }
    s2 = block_sum_(s2, red);
    {
        const float inv2 = 1.0f / s2;
#pragma unroll
        for (int i = 0; i < PER; ++i) lr[i] *= inv2;
    }

    // --- argmax(lr + g3), first-index tie-break ---
    float bm = -FLT_MAX; int bidx = 0x7fffffff;
#pragma unroll
    for (int i = 0; i < PER; ++i) {
        const float a = lr[i] + g3[i];
        const int v = tid + i * 256;
        if (a > bm || (a == bm && v < bidx)) { bm = a; bidx = v; }
    }
    redv[tid] = bm; redi[tid] = bidx; __syncthreads();
    for (int st = 128; st > 0; st >>= 1) {
        if (tid < st) {
            const float ov = redv[tid + st]; const int oi = redi[tid + st];
            if (ov > redv[tid] || (ov == redv[tid] && oi < redi[tid])) {
                redv[tid] = ov; redi[tid] = oi;
            }
        }
        __syncthreads();
    }
    const int widx = redi[0]; __syncthreads();
    {
        float* ohrow = ohs + ((size_t)b * (T_ + 1) + t) * V_;
#pragma unroll
        for (int i = 0; i < PER; ++i) {
            const int v = tid + i * 256;
            ohrow[v] = (v == widx) ? 1.0f : 0.0f;
        }
    }

    // --- log_prob = log_softmax(lr); entropy = -sum(log_prob * softmax(log_prob)) ---
    float m3 = -FLT_MAX;
#pragma unroll
    for (int i = 0; i < PER; ++i) m3 = fmaxf(m3, lr[i]);
    m3 = block_max_(m3, red);
    float s3 = 0.f;
#pragma unroll
    for (int i = 0; i < PER; ++i) s3 += expf(lr[i] - m3);
    s3 = block_sum_(s3, red);
    const float ls3 = logf(s3);
    const float inv3 = 1.0f / s3;
    float entp = 0.f;
    {
        float* lprow = lps + ((size_t)b * T_ + t) * V_;
#pragma unroll
        for (int i = 0; i < PER; ++i) {
            float lp = lr[i] - m3 - ls3;
            lp = fmaxf(lp, -FLT_MAX);   // clamp to finfo min
            lprow[tid + i * 256] = lp;
            const float p = expf(lr[i] - m3) * inv3;  // softmax(log_prob) == softmax(lr)
            entp -= lp * p;
        }
    }
    entp = block_sum_(entp, red);
    if (tid == 0) ent[(size_t)t * B_ + b] = entp;
}

// ---------------------------------------------------------------------------
// Init c0 = 0, e0 = broadcast(sos). One block per batch row.
// ---------------------------------------------------------------------------
__global__ void __launch_bounds__(256)
init_kernel(const float* __restrict__ sos, float* __restrict__ c0,
            float* __restrict__ e0) {
    const int b = blockIdx.x, tid = threadIdx.x;
#pragma unroll
    for (int i = 0; i < H_ / 256; ++i) c0[(size_t)b * H_ + tid + i * 256] = 0.f;
#pragma unroll
    for (int i = 0; i < E_ / 256; ++i) {
        const int j = tid + i * 256;
        e0[(size_t)b * E_ + j] = sos[j];
    }
}

// ---------------------------------------------------------------------------
// Append EOS one-hot row (index 0) at t = T for seq and ohs.
// ---------------------------------------------------------------------------
__global__ void __launch_bounds__(256)
eos_kernel(float* __restrict__ seq, float* __restrict__ ohs) {
    const int b = blockIdx.x, tid = threadIdx.x;
    float* sr = seq + ((size_t)b * (T_ + 1) + T_) * V_;
    float* orow = ohs + ((size_t)b * (T_ + 1) + T_) * V_;
#pragma unroll
    for (int i = 0; i < V_ / 256; ++i) {
        const int v = tid + i * 256;
        const float val = (v == 0) ? 1.0f : 0.0f;
        sr[v] = val;
        orow[v] = val;
    }
}

// ---------------------------------------------------------------------------
extern "C" void kernel_launch(void* const* d_in, const int* in_sizes, int n_in,
                              void* d_out, int out_size, void* d_ws, size_t ws_size,
                              hipStream_t stream) {
    const float* x       = (const float*)d_in[0];
    const float* W_agent = (const float*)d_in[1];
    const float* b_agent = (const float*)d_in[2];
    const float* sos     = (const float*)d_in[3];
    const float* W_ih    = (const float*)d_in[4];
    const float* W_hh    = (const float*)d_in[5];
    const float* b_ih    = (const float*)d_in[6];
    const float* b_hh    = (const float*)d_in[7];
    const float* W_out   = (const float*)d_in[8];
    const float* b_out   = (const float*)d_in[9];
    const float* W_emb   = (const float*)d_in[10];
    const float* b_emb   = (const float*)d_in[11];
    const float* u_con   = (const float*)d_in[12];
    const float* u_rel   = (const float*)d_in[13];
    const float* u_cat   = (const float*)d_in[14];

    float* out = (float*)d_out;
    float* seq = out;                                    // [B, T+1, V]
    float* ohs = seq + (size_t)B_ * (T_ + 1) * V_;       // [B, T+1, V]
    float* lps = ohs + (size_t)B_ * (T_ + 1) * V_;       // [B, T, V]
    float* ent = lps + (size_t)B_ * T_ * V_;             // [T, B]

    float* ws = (float*)d_ws;
    float* hbuf0  = ws;
    float* hbuf1  = hbuf0 + (size_t)B_ * H_;
    float* cbuf0  = hbuf1 + (size_t)B_ * H_;
    float* cbuf1  = cbuf0 + (size_t)B_ * H_;
    float* ebuf   = cbuf1 + (size_t)B_ * H_;             // [B, E]
    float* logits = ebuf + (size_t)B_ * E_;              // [B, V]
    float* hb[2] = {hbuf0, hbuf1};
    float* cb[2] = {cbuf0, cbuf1};

    // h0 = x @ W_agent^T + b_agent ; c0 = 0 ; e0 = sos
    gemm_bias_kernel<B_, H_, DIN_>
        <<<(B_ / 16) * (H_ / 64) / 8, 256, 0, stream>>>(x, DIN_, W_agent, b_agent, hb[0]);
    init_kernel<<<B_, 256, 0, stream>>>(sos, cb[0], ebuf);

    for (int t = 0; t < T_; ++t) {
        const int cur = t & 1, nxt = (t + 1) & 1;
        lstm_step_kernel<<<(B_ / 32) * (H_ / 16) / 8, 256, 0, stream>>>(
            ebuf, hb[cur], cb[cur], W_ih, W_hh, b_ih, b_hh, hb[nxt], cb[nxt]);
        gemm_bias_kernel<B_, V_, H_>
            <<<(B_ / 16) * (V_ / 64) / 8, 256, 0, stream>>>(hb[nxt], H_, W_out, b_out, logits);
        sample_kernel<<<B_, 256, 0, stream>>>(logits, u_con, u_rel, u_cat,
                                              seq, ohs, lps, ent, t);
        // e_new = sample_concrete @ W_emb^T + b_emb ; A rows live inside seq (lda = (T+1)*V)
        gemm_bias_kernel<B_, E_, V_>
            <<<(B_ / 16) * (E_ / 64) / 8, 256, 0, stream>>>(seq + (size_t)t * V_,
                                                            (T_ + 1) * V_, W_emb, b_emb, ebuf);
    }
    eos_kernel<<<B_, 256, 0, stream>>>(seq, ohs);
}